// ChangeFilter_65420941852735
// MI455X (gfx1250) — compile-verified
//
#include <hip/hip_runtime.h>
#include <math.h>

// Problem constants (fixed by the reference).
#define BN    32
#define CC    256
#define NPIX  4096          // H*W
#define NN    131072        // B*H*W
#define NE    524288
#define MHD   8
#define HD    32            // CC / MHD
#define INV_TEMP 10.0f
#define LN_EPS   1e-5f

typedef __attribute__((ext_vector_type(2))) float v2f;
typedef __attribute__((ext_vector_type(8))) float v8f;

// ---------------------------------------------------------------------------
// GEMM 1: xp[n,c] = sum_k img[b,k,pix] * Wp[k,c] + bp[c]
// One wave computes a 16(M) x 64(N) tile via 4 f32-WMMA accumulators.
// A-matrix layout (16x4 f32): lane l -> M = l&15, VGPR0/1 -> K = 2*(l>>4), +1
// B-matrix layout (4x16 f32): lane l -> N = l&15, VGPR0/1 -> K = 2*(l>>4), +1
// C/D (16x16 f32): lane l -> N = l&15, VGPR r -> M = r + 8*(l>>4)
// ---------------------------------------------------------------------------
__global__ __launch_bounds__(32)
void k_gemm_img(const float* __restrict__ img, const float* __restrict__ Wp,
                const float* __restrict__ bp, float* __restrict__ xp)
{
    const int tile = blockIdx.x;          // 0..8191   (M tiles of 16 nodes)
    const int ng   = blockIdx.y;          // 0..3      (64-column groups)
    const int m0   = tile * 16;
    const int b    = m0 >> 12;            // node tile never crosses batch (4096%16==0)
    const int pix0 = m0 & 4095;
    const int c0   = ng * 64;
    const int l    = threadIdx.x;
    const int mloc = l & 15;
    const int kb   = (l >> 4) << 1;

    v8f acc0 = {}, acc1 = {}, acc2 = {}, acc3 = {};
    const float* abase = img + (size_t)b * CC * NPIX + pix0 + mloc;

    for (int kk = 0; kk < CC; kk += 4) {
        const int k = kk + kb;
        v2f a;
        a.x = abase[(size_t)k       * NPIX];
        a.y = abase[(size_t)(k + 1) * NPIX];
        const float* br0 = Wp + (size_t)k * CC + c0 + mloc;
        const float* br1 = br0 + CC;
        v2f b0; b0.x = br0[0];  b0.y = br1[0];
        v2f b1; b1.x = br0[16]; b1.y = br1[16];
        v2f b2; b2.x = br0[32]; b2.y = br1[32];
        v2f b3; b3.x = br0[48]; b3.y = br1[48];
        acc0 = __builtin_amdgcn_wmma_f32_16x16x4_f32(false, a, false, b0, (short)0, acc0, false, false);
        acc1 = __builtin_amdgcn_wmma_f32_16x16x4_f32(false, a, false, b1, (short)0, acc1, false, false);
        acc2 = __builtin_amdgcn_wmma_f32_16x16x4_f32(false, a, false, b2, (short)0, acc2, false, false);
        acc3 = __builtin_amdgcn_wmma_f32_16x16x4_f32(false, a, false, b3, (short)0, acc3, false, false);
    }

    const int mh = (l >> 4) * 8;
    #pragma unroll
    for (int r = 0; r < 8; ++r) {
        const int row = m0 + mh + r;
        float* orow = xp + (size_t)row * CC + c0 + mloc;
        orow[0]  = acc0[r] + bp[c0 + mloc];
        orow[16] = acc1[r] + bp[c0 + 16 + mloc];
        orow[32] = acc2[r] + bp[c0 + 32 + mloc];
        orow[48] = acc3[r] + bp[c0 + 48 + mloc];
    }
}

// ---------------------------------------------------------------------------
// GEMM 2: xl = xp @ Wg   (row-major A, no bias — bias folded into agg init)
// ---------------------------------------------------------------------------
__global__ __launch_bounds__(32)
void k_gemm_x(const float* __restrict__ xp, const float* __restrict__ Wg,
              float* __restrict__ xl)
{
    const int tile = blockIdx.x;
    const int ng   = blockIdx.y;
    const int m0   = tile * 16;
    const int c0   = ng * 64;
    const int l    = threadIdx.x;
    const int mloc = l & 15;
    const int kb   = (l >> 4) << 1;

    v8f acc0 = {}, acc1 = {}, acc2 = {}, acc3 = {};
    const float* arow = xp + (size_t)(m0 + mloc) * CC;

    for (int kk = 0; kk < CC; kk += 4) {
        const int k = kk + kb;
        v2f a;
        a.x = arow[k];
        a.y = arow[k + 1];
        const float* br0 = Wg + (size_t)k * CC + c0 + mloc;
        const float* br1 = br0 + CC;
        v2f b0; b0.x = br0[0];  b0.y = br1[0];
        v2f b1; b1.x = br0[16]; b1.y = br1[16];
        v2f b2; b2.x = br0[32]; b2.y = br1[32];
        v2f b3; b3.x = br0[48]; b3.y = br1[48];
        acc0 = __builtin_amdgcn_wmma_f32_16x16x4_f32(false, a, false, b0, (short)0, acc0, false, false);
        acc1 = __builtin_amdgcn_wmma_f32_16x16x4_f32(false, a, false, b1, (short)0, acc1, false, false);
        acc2 = __builtin_amdgcn_wmma_f32_16x16x4_f32(false, a, false, b2, (short)0, acc2, false, false);
        acc3 = __builtin_amdgcn_wmma_f32_16x16x4_f32(false, a, false, b3, (short)0, acc3, false, false);
    }

    const int mh = (l >> 4) * 8;
    #pragma unroll
    for (int r = 0; r < 8; ++r) {
        const int row = m0 + mh + r;
        float* orow = xl + (size_t)row * CC + c0 + mloc;
        orow[0]  = acc0[r];
        orow[16] = acc1[r];
        orow[32] = acc2[r];
        orow[48] = acc3[r];
    }
}

// ---------------------------------------------------------------------------
// Degree / normalization (self-loop counted by init to 1.0)
// ---------------------------------------------------------------------------
__global__ void k_deg_init(float* __restrict__ deg)
{
    int i = blockIdx.x * blockDim.x + threadIdx.x;
    if (i < NN) deg[i] = 1.0f;
}

__global__ void k_deg_count(const long long* __restrict__ col, float* __restrict__ deg)
{
    int e = blockIdx.x * blockDim.x + threadIdx.x;
    if (e < NE) unsafeAtomicAdd(&deg[col[e]], 1.0f);
}

__global__ void k_dinv(const float* __restrict__ deg, float* __restrict__ dinv)
{
    int i = blockIdx.x * blockDim.x + threadIdx.x;
    if (i < NN) dinv[i] = rsqrtf(deg[i]);   // deg >= 1 always (self-loop)
}

// agg[n,c] = xl[n,c] * dinv[n]^2 (self loop) + bg[c]
__global__ void k_agg_init(const float* __restrict__ xl, const float* __restrict__ dinv,
                           const float* __restrict__ bg, float* __restrict__ agg)
{
    int idx = blockIdx.x * blockDim.x + threadIdx.x;   // < NN*CC = 33.5M (fits int)
    int n = idx >> 8;
    int c = idx & 255;
    float di = dinv[n];
    agg[idx] = xl[idx] * di * di + bg[c];
}

// One wave per edge: agg[col] += xl[row] * dinv[row]*dinv[col], 8 channels/lane
__global__ __launch_bounds__(256)
void k_edge(const long long* __restrict__ row, const long long* __restrict__ col,
            const float* __restrict__ xl, const float* __restrict__ dinv,
            float* __restrict__ agg)
{
    int e    = (blockIdx.x * 256 + threadIdx.x) >> 5;
    int lane = threadIdx.x & 31;
    if (e >= NE) return;
    long long r = row[e];
    long long cn = col[e];
    float nrm = dinv[r] * dinv[cn];
    const float* src = xl  + (size_t)r  * CC;
    float*       dst = agg + (size_t)cn * CC;
    #pragma unroll
    for (int j = 0; j < 8; ++j) {
        int c = lane + j * 32;
        unsafeAtomicAdd(&dst[c], src[c] * nrm);
    }
}

// ---------------------------------------------------------------------------
// Epilogue: g = GELU(agg) + xp ; LayerNorm over C -> img_graph (d_out part 1)
// One wave per node row, 8 channels/lane, wave-shuffle reductions (wave32).
// ---------------------------------------------------------------------------
__global__ __launch_bounds__(256)
void k_ln(const float* __restrict__ agg, const float* __restrict__ xp,
          const float* __restrict__ gamma, const float* __restrict__ beta,
          float* __restrict__ outg)
{
    int wave = threadIdx.x >> 5;
    int lane = threadIdx.x & 31;
    size_t n = (size_t)blockIdx.x * 8 + wave;
    const float* grow = agg + n * CC;
    const float* xrow = xp  + n * CC;

    float t[8];
    float s = 0.0f;
    #pragma unroll
    for (int j = 0; j < 8; ++j) {
        int c = lane + j * 32;
        float gv = grow[c];
        float ge = 0.5f * gv * (1.0f + erff(gv * 0.70710678118654752f)); // exact GELU
        t[j] = ge + xrow[c];
        s += t[j];
    }
    #pragma unroll
    for (int m = 16; m >= 1; m >>= 1) s += __shfl_xor(s, m, 32);
    float mu = s * (1.0f / 256.0f);

    float vs = 0.0f;
    #pragma unroll
    for (int j = 0; j < 8; ++j) { float d = t[j] - mu; vs += d * d; }
    #pragma unroll
    for (int m = 16; m >= 1; m >>= 1) vs += __shfl_xor(vs, m, 32);
    float rinv = rsqrtf(vs * (1.0f / 256.0f) + LN_EPS);

    float* orow = outg + n * CC;
    #pragma unroll
    for (int j = 0; j < 8; ++j) {
        int c = lane + j * 32;
        orow[c] = (t[j] - mu) * rinv * gamma[c] + beta[c];
    }
}

// ---------------------------------------------------------------------------
// Adjacency: logits = (q_h . k) / T, softmax over N=4096 per (batch, head).
// One 256-thread block per (b, head); 16 logits per thread in registers.
// ---------------------------------------------------------------------------
__global__ __launch_bounds__(256)
void k_adj(const float* __restrict__ ig, const float* __restrict__ dq,
           float* __restrict__ adj)
{
    int bm  = blockIdx.x;          // 0..255  = b*MHD + head
    int b   = bm >> 3;
    int mh  = bm & 7;
    int tid = threadIdx.x;

    __shared__ float qs[HD];
    __shared__ float red[256];
    if (tid < HD) qs[tid] = dq[mh * HD + tid];
    __syncthreads();

    float lg[16];
    float lmax = -3.402823466e+38f;
    #pragma unroll
    for (int i = 0; i < 16; ++i) {
        int n = i * 256 + tid;
        const float* krow = ig + ((size_t)(b * NPIX + n)) * CC + mh * HD;
        float d = 0.0f;
        #pragma unroll
        for (int c = 0; c < HD; ++c) d += qs[c] * krow[c];
        lg[i] = d * INV_TEMP;
        lmax = fmaxf(lmax, lg[i]);
    }

    red[tid] = lmax;
    __syncthreads();
    for (int s = 128; s > 0; s >>= 1) {
        if (tid < s) red[tid] = fmaxf(red[tid], red[tid + s]);
        __syncthreads();
    }
    float gmax = red[0];
    __syncthreads();

    float ls = 0.0f;
    #pragma unroll
    for (int i = 0; i < 16; ++i) { lg[i] = expf(lg[i] - gmax); ls += lg[i]; }
    red[tid] = ls;
    __syncthreads();
    for (int s = 128; s > 0; s >>= 1) {
        if (tid < s) red[tid] += red[tid + s];
        __syncthreads();
    }
    float inv = 1.0f / red[0];

    float* orow = adj + (size_t)bm * NPIX;
    #pragma unroll
    for (int i = 0; i < 16; ++i) orow[i * 256 + tid] = lg[i] * inv;
}

// ---------------------------------------------------------------------------
extern "C" void kernel_launch(void* const* d_in, const int* in_sizes, int n_in,
                              void* d_out, int out_size, void* d_ws, size_t ws_size,
                              hipStream_t stream)
{
    const float*     img   = (const float*)d_in[0];
    const long long* edges = (const long long*)d_in[1];   // int64 [2, E]
    const float*     Wp    = (const float*)d_in[2];
    const float*     bp    = (const float*)d_in[3];
    const float*     Wg    = (const float*)d_in[4];
    const float*     bg    = (const float*)d_in[5];
    const float*     gamma = (const float*)d_in[6];
    const float*     beta  = (const float*)d_in[7];
    const float*     dq    = (const float*)d_in[8];
    (void)in_sizes; (void)n_in; (void)out_size; (void)ws_size;

    float* outg = (float*)d_out;                 // img_graph [NN, CC]
    float* adj  = outg + (size_t)NN * CC;        // adj [B*MH, NPIX]

    float* xp   = (float*)d_ws;                  // [NN, CC]
    float* xl   = xp  + (size_t)NN * CC;         // [NN, CC]
    float* agg  = xl  + (size_t)NN * CC;         // [NN, CC]
    float* deg  = agg + (size_t)NN * CC;         // [NN]
    float* dinv = deg + NN;                      // [NN]

    const long long* row = edges;
    const long long* col = edges + NE;

    dim3 gGemm(NN / 16, 4);
    k_gemm_img<<<gGemm, 32, 0, stream>>>(img, Wp, bp, xp);
    k_gemm_x  <<<gGemm, 32, 0, stream>>>(xp, Wg, xl);

    k_deg_init <<<NN / 256, 256, 0, stream>>>(deg);
    k_deg_count<<<NE / 256, 256, 0, stream>>>(col, deg);
    k_dinv     <<<NN / 256, 256, 0, stream>>>(deg, dinv);

    k_agg_init<<<(NN * CC) / 256, 256, 0, stream>>>(xl, dinv, bg, agg);
    k_edge    <<<(NE * 32) / 256, 256, 0, stream>>>(row, col, xl, dinv, agg);

    k_ln <<<NN / 8, 256, 0, stream>>>(agg, xp, gamma, beta, outg);
    k_adj<<<BN * MHD, 256, 0, stream>>>(outg, dq, adj);
}